// GraphGRUCell_28269474742856
// MI455X (gfx1250) — compile-verified
//
#include <hip/hip_runtime.h>
#include <hip/hip_bf16.h>

#define Bdim 2048
#define Hdim 256

typedef __attribute__((ext_vector_type(2))) float v2f;
typedef __attribute__((ext_vector_type(4))) float v4f;
typedef __attribute__((ext_vector_type(8))) float v8f;

__device__ __forceinline__ float sigmoidf(float x) {
    return 1.0f / (1.0f + expf(-x));
}

// ---------------------------------------------------------------------------
// Kernel 1: batched matvec  out[m][b][i] = dot(W_m[b,i,:], x[b,:])
// grid = (B, 3), block = 256 (8 waves). Streams 1.5 GB of W with NT b128 loads.
// Each wave handles 32 rows; per-lane x slice held in registers; wave32 reduce.
// ---------------------------------------------------------------------------
__global__ void __launch_bounds__(256)
bmm3_kernel(const float* __restrict__ Wz, const float* __restrict__ Wr,
            const float* __restrict__ Wh, const float* __restrict__ x,
            float* __restrict__ out /* [3][B][H] */) {
    const int b = blockIdx.x;
    const int m = blockIdx.y;
    const float* __restrict__ W = (m == 0) ? Wz : (m == 1) ? Wr : Wh;

    const int lane = threadIdx.x & 31;
    const int wave = threadIdx.x >> 5;

    const float* __restrict__ xb = x + (size_t)b * Hdim;
    // Each lane owns x[4l..4l+3] and x[128+4l..128+4l+3]; same for every row.
    const v4f xa = *(const v4f*)(xb + 4 * lane);
    const v4f xc = *(const v4f*)(xb + 128 + 4 * lane);

    const float* __restrict__ Wb = W + (size_t)b * Hdim * Hdim;
    float* __restrict__ ob = out + (size_t)m * Bdim * Hdim + (size_t)b * Hdim;

#pragma unroll 2
    for (int r = 0; r < 32; ++r) {
        const int i = wave * 32 + r;
        const float* __restrict__ row = Wb + (size_t)i * Hdim;
        // Streaming data: non-temporal so 1.5GB doesn't thrash L2.
        v4f wa = __builtin_nontemporal_load((const v4f*)(row + 4 * lane));
        v4f wc = __builtin_nontemporal_load((const v4f*)(row + 128 + 4 * lane));
        float acc = wa.x * xa.x + wa.y * xa.y + wa.z * xa.z + wa.w * xa.w
                  + wc.x * xc.x + wc.y * xc.y + wc.z * xc.z + wc.w * xc.w;
        // wave32 tree reduction
        acc += __shfl_xor(acc, 16, 32);
        acc += __shfl_xor(acc, 8, 32);
        acc += __shfl_xor(acc, 4, 32);
        acc += __shfl_xor(acc, 2, 32);
        acc += __shfl_xor(acc, 1, 32);
        if (lane == 0) ob[i] = acc;
    }
}

// ---------------------------------------------------------------------------
// Kernel 2: dense GEMM  out[B,H] = X[B,H] @ W^T  (W is [H,H] row-major, i.e.
// out[m,n] = sum_k X[m,k] * W[n,k]) using V_WMMA_F32_16X16X4_F32.
// grid = (B/16, H/16); one wave (block of 32) per 16x16 output tile.
//
// f32 WMMA operand layouts (ISA 7.12.2):
//   A (16x4, MxK): lanes 0-15 -> M=lane {VGPR0:K=k0, VGPR1:K=k0+1},
//                  lanes 16-31 -> M=lane-16 {VGPR0:K=k0+2, VGPR1:K=k0+3}
//   B (4x16, KxN): lanes 0-15 -> N=lane {VGPR0:K=k0, VGPR1:K=k0+1},
//                  lanes 16-31 -> N=lane-16 {VGPR0:K=k0+2, VGPR1:K=k0+3}
//   C/D (16x16):   VGPR r: lanes 0-15 -> (M=r, N=lane),
//                          lanes 16-31 -> (M=8+r, N=lane-16)
// ---------------------------------------------------------------------------
__global__ void __launch_bounds__(32)
gemm_wmma_f32(const float* __restrict__ X, const float* __restrict__ Wt,
              float* __restrict__ out) {
    const int m0   = blockIdx.x * 16;
    const int n0   = blockIdx.y * 16;
    const int lane = threadIdx.x;      // 0..31
    const int half = lane >> 4;        // 0 or 1
    const int lr   = lane & 15;

    // Row pointers as float2: element [j] = {row[2j], row[2j+1]} (16B-aligned rows)
    const v2f* __restrict__ xrow = (const v2f*)(X  + (size_t)(m0 + lr) * Hdim);
    const v2f* __restrict__ wrow = (const v2f*)(Wt + (size_t)(n0 + lr) * Hdim);

    v8f acc = {};
#pragma unroll 4
    for (int k2 = 0; k2 < Hdim / 2; k2 += 2) {  // k0 = 2*k2, steps of K=4
        v2f a = xrow[k2 + half];   // {X[m0+lr, k0+2h], X[m0+lr, k0+2h+1]}
        v2f bb = wrow[k2 + half];  // {W[n0+lr, k0+2h], W[n0+lr, k0+2h+1]}
        acc = __builtin_amdgcn_wmma_f32_16x16x4_f32(
            /*neg_a=*/false, a, /*neg_b=*/false, bb,
            /*c_mod=*/(short)0, acc, /*reuse_a=*/false, /*reuse_b=*/false);
    }

#pragma unroll
    for (int r = 0; r < 8; ++r) {
        const int mrow = m0 + r + (half ? 8 : 0);
        out[(size_t)mrow * Hdim + n0 + lr] = acc[r];
    }
}

// ---------------------------------------------------------------------------
// Kernel 3: gates  z = sig(Wz_x + Uz), r = sig(Wr_x + Ur),
//                  g = sig(pf @ pg_w^T + pg_b), rp = r * h_prev
// ---------------------------------------------------------------------------
__global__ void __launch_bounds__(256)
gates_kernel(const float* __restrict__ Wz_x, const float* __restrict__ Wr_x,
             const float* __restrict__ Uz, const float* __restrict__ Ur,
             const float* __restrict__ hp, const float* __restrict__ pf,
             const float* __restrict__ pg_w, const float* __restrict__ pg_b,
             float* __restrict__ z_t, float* __restrict__ g_t,
             float* __restrict__ rp) {
    const int idx = blockIdx.x * 256 + threadIdx.x;   // < B*H
    const int b = idx >> 8;
    const int h = idx & (Hdim - 1);

    const float z = sigmoidf(Wz_x[idx] + Uz[idx]);
    const float r = sigmoidf(Wr_x[idx] + Ur[idx]);
    const float gl = pf[b * 3 + 0] * pg_w[h * 3 + 0]
                   + pf[b * 3 + 1] * pg_w[h * 3 + 1]
                   + pf[b * 3 + 2] * pg_w[h * 3 + 2] + pg_b[h];
    z_t[idx] = z;
    g_t[idx] = sigmoidf(gl);
    rp[idx]  = r * hp[idx];
}

// ---------------------------------------------------------------------------
// Kernel 4: final fuse
//   cand = tanh(Wh_x + Uh); comb = (1-z)h + z*cand; ht = (1-g)h + g*comb
//   out  = visit_active[b] ? ht : h
// ---------------------------------------------------------------------------
__global__ void __launch_bounds__(256)
final_kernel(const float* __restrict__ Wh_x, const float* __restrict__ Uh,
             const float* __restrict__ z_t, const float* __restrict__ g_t,
             const float* __restrict__ hp, const unsigned char* __restrict__ act,
             float* __restrict__ out) {
    const int idx = blockIdx.x * 256 + threadIdx.x;   // < B*H
    const int b = idx >> 8;

    const float h0   = hp[idx];
    const float cand = tanhf(Wh_x[idx] + Uh[idx]);
    const float z = z_t[idx];
    const float g = g_t[idx];
    const float comb = (1.0f - z) * h0 + z * cand;
    const float ht   = (1.0f - g) * h0 + g * comb;
    out[idx] = act[b] ? ht : h0;
}

extern "C" void kernel_launch(void* const* d_in, const int* in_sizes, int n_in,
                              void* d_out, int out_size, void* d_ws, size_t ws_size,
                              hipStream_t stream) {
    const float* graph_state = (const float*)d_in[0];   // [B,H]
    const float* prev_hidden = (const float*)d_in[1];   // [B,H]
    const float* W_z = (const float*)d_in[2];           // [B,H,H]
    const float* W_r = (const float*)d_in[3];           // [B,H,H]
    const float* W_h = (const float*)d_in[4];           // [B,H,H]
    const float* progress = (const float*)d_in[5];      // [B,3]
    const unsigned char* visit_active = (const unsigned char*)d_in[6]; // [B] bool
    const float* Uz_w = (const float*)d_in[7];          // [H,H]
    const float* Ur_w = (const float*)d_in[8];          // [H,H]
    const float* Uh_w = (const float*)d_in[9];          // [H,H]
    const float* pg_w = (const float*)d_in[10];         // [H,3]
    const float* pg_b = (const float*)d_in[11];         // [H]
    float* out = (float*)d_out;

    const size_t BH = (size_t)Bdim * Hdim;
    float* ws   = (float*)d_ws;
    float* Wz_x = ws + 0 * BH;
    float* Wr_x = ws + 1 * BH;
    float* Wh_x = ws + 2 * BH;
    float* Uz   = ws + 3 * BH;
    float* Ur   = ws + 4 * BH;
    float* Uh   = ws + 5 * BH;
    float* z_t  = ws + 6 * BH;
    float* g_t  = ws + 7 * BH;
    float* rp   = ws + 8 * BH;

    // 1) Streaming batched matvecs (bandwidth-dominant: 1.5 GB of W, once).
    bmm3_kernel<<<dim3(Bdim, 3), 256, 0, stream>>>(W_z, W_r, W_h, graph_state, ws);

    // 2) Uz = prev_hidden @ Uz_w^T ; Ur = prev_hidden @ Ur_w^T  (WMMA f32)
    dim3 ggrid(Bdim / 16, Hdim / 16);
    gemm_wmma_f32<<<ggrid, 32, 0, stream>>>(prev_hidden, Uz_w, Uz);
    gemm_wmma_f32<<<ggrid, 32, 0, stream>>>(prev_hidden, Ur_w, Ur);

    // 3) z, r, g gates + rp = r * prev_hidden
    gates_kernel<<<(unsigned)(BH / 256), 256, 0, stream>>>(
        Wz_x, Wr_x, Uz, Ur, prev_hidden, progress, pg_w, pg_b, z_t, g_t, rp);

    // 4) Uh = (r * prev_hidden) @ Uh_w^T  (WMMA f32)
    gemm_wmma_f32<<<ggrid, 32, 0, stream>>>(rp, Uh_w, Uh);

    // 5) final fuse
    final_kernel<<<(unsigned)(BH / 256), 256, 0, stream>>>(
        Wh_x, Uh, z_t, g_t, prev_hidden, visit_active, out);
}